// RelTransformerQ_23287312679396
// MI455X (gfx1250) — compile-verified
//
#include <hip/hip_runtime.h>
#include <hip/hip_bf16.h>
#include <math.h>

// ---------------------------------------------------------------------------
// RelTransformerQ for MI455X (gfx1250, wave32).
//
// Shapes: B=256, T=128, SD=8, AD=2, H=3, K=4, HID=256.
// Key transform: rel@Wk.T is linear in s, so keys[b,t,j] = ks[idx]-ks[t]
// with ks = s4@Wk.T (B,T,12). Avoids the reference's ~800MB keys/vals
// tensors entirely. Softmax weights sum to 1, so the val-side collapses to
// (sum_j w*vs[idx]) - vs[t].
//
// Big GEMMs (32768x24x256 and 32768x256x256) run on V_WMMA_F32_16X16X4_F32
// (fp32 in / fp32 accumulate, matching reference precision), with 1x4
// register blocking (16x64 strip per wave) and explicit register
// double-buffering: step k+1's A/B fragments load while step k's 4 WMMAs
// execute, so VMEM latency hides behind the matrix pipe.
// ---------------------------------------------------------------------------

typedef __attribute__((ext_vector_type(2))) float v2f;
typedef __attribute__((ext_vector_type(8))) float v8f;

#define B_   256
#define T_   128
#define SD_  8
#define AD_  2
#define H_   3
#define KD_  4          // head dim
#define HID_ 256
#define QDIM 10         // SD+AD
#define HK   12         // H*KD
#define X24  24         // 12 attn + 10 q_state, padded 22 -> 24 for K%4==0
#define ROWS (B_ * T_)  // 32768
#define NBLK 4          // N-tiles (16 cols each) per wave

// workspace layout (float offsets)
static constexpr size_t OFF_Q   = 0;
static constexpr size_t OFF_KS  = OFF_Q  + (size_t)ROWS * HK;
static constexpr size_t OFF_VS  = OFF_KS + (size_t)ROWS * HK;
static constexpr size_t OFF_X24 = OFF_VS + (size_t)ROWS * HK;
static constexpr size_t OFF_W1P = OFF_X24 + (size_t)ROWS * X24;
static constexpr size_t OFF_Y1  = OFF_W1P + (size_t)HID_ * X24;
static constexpr size_t OFF_Y2  = OFF_Y1  + (size_t)ROWS * HID_;

// ---------------------------------------------------------------------------
// Kernel 1: per-(b,t) projections  q = Wq@[s;a],  ks = Wk@s4,  vs = Wv@s4
// ---------------------------------------------------------------------------
__global__ void prep_kernel(const float* __restrict__ state,
                            const float* __restrict__ action,
                            const float* __restrict__ Wk,
                            const float* __restrict__ Wq,
                            const float* __restrict__ Wv,
                            float* __restrict__ q,
                            float* __restrict__ ks,
                            float* __restrict__ vs) {
  const int idx = blockIdx.x * blockDim.x + threadIdx.x;  // < ROWS exactly
  const float* st = state  + (size_t)idx * SD_;
  const float* ac = action + (size_t)idx * AD_;

  float qs[QDIM];
#pragma unroll
  for (int d = 0; d < SD_; ++d) qs[d] = st[d];
  qs[8] = ac[0];
  qs[9] = ac[1];
  const float s4[4] = {qs[0], qs[1], qs[2], qs[3]};

  float* qp = q  + (size_t)idx * HK;
  float* kp = ks + (size_t)idx * HK;
  float* vp = vs + (size_t)idx * HK;
#pragma unroll
  for (int r = 0; r < HK; ++r) {
    float aq = 0.f;
#pragma unroll
    for (int d = 0; d < QDIM; ++d) aq += Wq[r * QDIM + d] * qs[d];
    qp[r] = aq;
    float ak = 0.f, av = 0.f;
#pragma unroll
    for (int c = 0; c < 4; ++c) {
      ak += Wk[r * 4 + c] * s4[c];
      av += Wv[r * 4 + c] * s4[c];
    }
    kp[r] = ak;
    vp[r] = av;
  }
}

// ---------------------------------------------------------------------------
// Kernel 2: zero-pad W1 (256 x 22) -> W1p (256 x 24)
// ---------------------------------------------------------------------------
__global__ void pad_w1_kernel(const float* __restrict__ W1,
                              float* __restrict__ W1p) {
  const int i = blockIdx.x * blockDim.x + threadIdx.x;
  if (i >= HID_ * X24) return;
  const int row = i / X24, col = i - row * X24;
  W1p[i] = (col < 22) ? W1[row * 22 + col] : 0.0f;
}

// ---------------------------------------------------------------------------
// Kernel 3: attention. One block per b; ks/vs for the batch staged in LDS.
// One thread per (t,h) row runs an online softmax over the 127 j != t.
// Writes x24 row = [attn(12) | state(8) | action(2) | 0 0].
// ---------------------------------------------------------------------------
__global__ void attn_kernel(const float* __restrict__ q,
                            const float* __restrict__ ks,
                            const float* __restrict__ vs,
                            const float* __restrict__ state,
                            const float* __restrict__ action,
                            float* __restrict__ x24) {
  __shared__ float ks_s[T_ * HK];  // 6 KB
  __shared__ float vs_s[T_ * HK];  // 6 KB

  const int b   = blockIdx.x;
  const int tid = threadIdx.x;  // 384 = T_*H_
  const float* ksb = ks + (size_t)b * T_ * HK;
  const float* vsb = vs + (size_t)b * T_ * HK;
  for (int i = tid; i < T_ * HK; i += T_ * H_) {
    ks_s[i] = ksb[i];
    vs_s[i] = vsb[i];
  }
  __syncthreads();

  const int t = tid / H_;
  const int h = tid - t * H_;
  const size_t row = (size_t)b * T_ + t;

  float qv[4], kst[4], vst[4];
#pragma unroll
  for (int k = 0; k < 4; ++k) {
    qv[k]  = q[row * HK + h * KD_ + k];
    kst[k] = ks_s[t * HK + h * KD_ + k];
    vst[k] = vs_s[t * HK + h * KD_ + k];
  }

  float m = -3.4e38f, l = 0.f;
  float a0 = 0.f, a1 = 0.f, a2 = 0.f, a3 = 0.f;
  for (int j = 0; j < T_; ++j) {
    if (j == t) continue;  // excluded index (idx skips j==t)
    const float* kj = ks_s + j * HK + h * KD_;
    float sc = (qv[0] * (kj[0] - kst[0]) + qv[1] * (kj[1] - kst[1]) +
                qv[2] * (kj[2] - kst[2]) + qv[3] * (kj[3] - kst[3])) * 0.5f;
    const float mn  = fmaxf(m, sc);
    const float cor = __expf(m - mn);
    const float p   = __expf(sc - mn);
    l = l * cor + p;
    const float* vj = vs_s + j * HK + h * KD_;
    a0 = a0 * cor + p * vj[0];
    a1 = a1 * cor + p * vj[1];
    a2 = a2 * cor + p * vj[2];
    a3 = a3 * cor + p * vj[3];
    m = mn;
  }
  const float inv = 1.0f / l;
  float* xr = x24 + row * X24;
  xr[h * KD_ + 0] = a0 * inv - vst[0];   // sum(w)=1 -> subtract vs[t] once
  xr[h * KD_ + 1] = a1 * inv - vst[1];
  xr[h * KD_ + 2] = a2 * inv - vst[2];
  xr[h * KD_ + 3] = a3 * inv - vst[3];
  if (h == 0) {
    const float* st = state  + row * SD_;
    const float* ac = action + row * AD_;
#pragma unroll
    for (int d = 0; d < SD_; ++d) xr[HK + d] = st[d];
    xr[20] = ac[0];
    xr[21] = ac[1];
    xr[22] = 0.0f;   // padding columns (W1p is zero there too)
    xr[23] = 0.0f;
  }
}

// ---------------------------------------------------------------------------
// Kernels 4/5: fp32 WMMA GEMM  Y = act(X @ W.T + bias)
// X: M x K row-major, W: N x K row-major (so B-matrix = W.T is K x N).
// One wave per 16x64 output strip (NBLK=4 tiles). Register double-buffered:
// fragments of step k+1 are issued before the WMMAs of step k. B columns of
// a strip are addressed as one base + compile-time immediate offsets
// (c*16*K floats <= 48KB, fits 24-bit IOFFSET).
// Lane layouts per CDNA5 ISA 7.12.2:
//   A 16x4 : lane(0-15)=rows, vgpr j holds K = j + 2*(lane>>4)
//   B 4x16 : lane&15 = col,   vgpr j holds K = j + 2*(lane>>4)
//   C 16x16: vgpr v, lane l -> row v + 8*(l>>4), col l&15
// ---------------------------------------------------------------------------
template <int K, bool RELU>
__global__ void gemm_wmma_kernel(const float* __restrict__ X,
                                 const float* __restrict__ W,
                                 const float* __restrict__ bias,
                                 float* __restrict__ Y,
                                 int N) {
  const int lane  = threadIdx.x & 31;
  const int wave  = threadIdx.x >> 5;
  const int hi16  = lane >> 4;
  const int lr    = lane & 15;

  const int nstrips = N >> 6;                              // 64-wide strips
  const int strip   = blockIdx.x * (blockDim.x >> 5) + wave;
  const int mtile   = strip / nstrips;
  const int nstrip  = strip - mtile * nstrips;
  const int m0 = mtile << 4;
  const int n0 = nstrip << 6;

  const float* Xr  = X + (size_t)(m0 + lr) * K + 2 * hi16; // A row for lane
  const float* Wr0 = W + (size_t)(n0 + lr) * K + 2 * hi16; // B col 0 for lane

  v8f acc[NBLK];
#pragma unroll
  for (int c = 0; c < NBLK; ++c) {
    const float bv = bias[n0 + c * 16 + lr];               // bias per column
#pragma unroll
    for (int i = 0; i < 8; ++i) acc[c][i] = bv;
  }

  // prologue: fragments for k-step 0
  v2f a_cur = *(const v2f*)(Xr);
  v2f b_cur[NBLK];
#pragma unroll
  for (int c = 0; c < NBLK; ++c)
    b_cur[c] = *(const v2f*)(Wr0 + (size_t)c * 16 * K);    // imm offset

#pragma unroll 4
  for (int kb = 0; kb < K; kb += 4) {
    // issue next step's loads first (distinct regs -> one clause, no
    // loadcnt-0 wait in front of every WMMA)
    v2f a_nxt = {};
    v2f b_nxt[NBLK] = {};
    if (kb + 4 < K) {
      a_nxt = *(const v2f*)(Xr + kb + 4);
#pragma unroll
      for (int c = 0; c < NBLK; ++c)
        b_nxt[c] = *(const v2f*)(Wr0 + (size_t)c * 16 * K + kb + 4);
    }
    // 4 WMMAs on current fragments
#pragma unroll
    for (int c = 0; c < NBLK; ++c) {
      acc[c] = __builtin_amdgcn_wmma_f32_16x16x4_f32(
          /*neg_a=*/false, a_cur, /*neg_b=*/false, b_cur[c],
          /*c_mod=*/(short)0, acc[c], /*reuse_a=*/false, /*reuse_b=*/false);
    }
    a_cur = a_nxt;
#pragma unroll
    for (int c = 0; c < NBLK; ++c) b_cur[c] = b_nxt[c];
  }

#pragma unroll
  for (int c = 0; c < NBLK; ++c) {
    float* Yr = Y + (size_t)m0 * N + n0 + c * 16 + lr;
#pragma unroll
    for (int v = 0; v < 8; ++v) {
      float r = acc[c][v];
      if (RELU) r = fmaxf(r, 0.0f);
      Yr[(size_t)(v + 8 * hi16) * N] = r;
    }
  }
}

// ---------------------------------------------------------------------------
// Kernel 6: out = Y2 @ Wout.T + bout. One wave per row, shuffle reduction.
// ---------------------------------------------------------------------------
__global__ void out_kernel(const float* __restrict__ Y2,
                           const float* __restrict__ Wout,
                           const float* __restrict__ bout,
                           float* __restrict__ out) {
  const int wave = threadIdx.x >> 5;
  const int lane = threadIdx.x & 31;
  const int row  = blockIdx.x * (blockDim.x >> 5) + wave;
  const float* yr = Y2 + (size_t)row * HID_;
  float s = 0.f;
#pragma unroll
  for (int i = 0; i < HID_ / 32; ++i) {
    const int c = lane + i * 32;
    s += yr[c] * Wout[c];
  }
#pragma unroll
  for (int off = 16; off > 0; off >>= 1) s += __shfl_xor(s, off, 32);
  if (lane == 0) out[row] = s + bout[0];
}

// ---------------------------------------------------------------------------
// Launch
// ---------------------------------------------------------------------------
extern "C" void kernel_launch(void* const* d_in, const int* in_sizes, int n_in,
                              void* d_out, int out_size, void* d_ws, size_t ws_size,
                              hipStream_t stream) {
  const float* state  = (const float*)d_in[0];
  const float* action = (const float*)d_in[1];
  const float* Wk     = (const float*)d_in[2];
  const float* Wq     = (const float*)d_in[3];
  const float* Wv     = (const float*)d_in[4];
  const float* W1     = (const float*)d_in[5];
  const float* b1     = (const float*)d_in[6];
  const float* W2     = (const float*)d_in[7];
  const float* b2     = (const float*)d_in[8];
  const float* Wout   = (const float*)d_in[9];
  const float* bout   = (const float*)d_in[10];
  float* out = (float*)d_out;

  float* w   = (float*)d_ws;
  float* q   = w + OFF_Q;
  float* ks  = w + OFF_KS;
  float* vs  = w + OFF_VS;
  float* x24 = w + OFF_X24;
  float* W1p = w + OFF_W1P;
  float* Y1  = w + OFF_Y1;
  float* Y2  = w + OFF_Y2;

  // 1) projections
  prep_kernel<<<ROWS / 256, 256, 0, stream>>>(state, action, Wk, Wq, Wv, q, ks, vs);
  // 2) pad W1 to K=24
  pad_w1_kernel<<<(HID_ * X24 + 255) / 256, 256, 0, stream>>>(W1, W1p);
  // 3) attention -> x24
  attn_kernel<<<B_, T_ * H_, 0, stream>>>(q, ks, vs, state, action, x24);
  // 4) Y1 = relu(x24 @ W1p.T + b1)   [M=32768, K=24, N=256]
  {
    const int waves = (ROWS / 16) * (HID_ / 64);  // 16x64 strips
    gemm_wmma_kernel<X24, true><<<waves / 8, 256, 0, stream>>>(x24, W1p, b1, Y1, HID_);
  }
  // 5) Y2 = relu(Y1 @ W2.T + b2)     [M=32768, K=256, N=256]
  {
    const int waves = (ROWS / 16) * (HID_ / 64);
    gemm_wmma_kernel<HID_, true><<<waves / 8, 256, 0, stream>>>(Y1, W2, b2, Y2, HID_);
  }
  // 6) out = Y2 @ Wout.T + bout
  out_kernel<<<ROWS / 8, 256, 0, stream>>>(Y2, Wout, bout, out);
}